// StratifiedMaxPooling_66314295050401
// MI455X (gfx1250) — compile-verified
//
#include <hip/hip_runtime.h>
#include <stdint.h>

// ---------------------------------------------------------------------------
// Stratified (label-grouped) max pooling.
//   values: [B, N] f32, labels: [N] i32, out: [B, C] f32
//   out[b,c] = max_{j : labels[j]==c} values[b,j]   (empty segment -> -inf)
//
// Pure bandwidth-bound (102.4 MB stream @ 23.3 TB/s ~= 4.5 us floor).
// WMMA is inapplicable: segment-max is a (max,*)-semiring reduction, not +,x.
// CDNA5 paths used instead:
//   * global_load_b128 with NT temporal hint for the one-pass values stream
//   * async global->LDS copy (ASYNCcnt) to stage labels, reused by 16 rows
//   * ds_max_u32 LDS atomics on an order-preserving u32 encoding of f32
//   * global_atomic_max_u32 to merge per-block partials
// ---------------------------------------------------------------------------

#define TPB 256
#define RT  16   // rows per block tile

typedef float v4f __attribute__((ext_vector_type(4)));
typedef int   v4i __attribute__((ext_vector_type(4)));

#define AS1 __attribute__((address_space(1)))
#define AS3 __attribute__((address_space(3)))

#if defined(__has_builtin)
#  if __has_builtin(__builtin_amdgcn_global_load_async_to_lds_b32) && \
      __has_builtin(__builtin_amdgcn_s_wait_asynccnt)
#    define HAVE_ASYNC_LDS 1
#  endif
#endif
#ifndef HAVE_ASYNC_LDS
#  define HAVE_ASYNC_LDS 0
#endif

// Order-preserving f32 -> u32 map: for floats a<b  =>  enc(a)<enc(b) (unsigned)
__device__ __forceinline__ unsigned enc_f32(float f) {
  unsigned u = __float_as_uint(f);
  return u ^ (unsigned)(((int)u >> 31) | (int)0x80000000);
}

__device__ __forceinline__ float dec_u32(unsigned e) {
  unsigned m = ((int)e < 0) ? 0x80000000u : 0xFFFFFFFFu;
  return __uint_as_float(e ^ m);
}

__global__ void __launch_bounds__(TPB)
strat_init_kernel(unsigned* __restrict__ acc, int n) {
  int i = blockIdx.x * TPB + threadIdx.x;
  if (i < n) acc[i] = 0u;   // 0 < enc(x) for every real float x
}

__global__ void __launch_bounds__(TPB)
strat_main_kernel(const float* __restrict__ values,
                  const int*   __restrict__ labels,
                  unsigned*    __restrict__ acc,
                  int N, int B, int C, int CP, int cpb) {
  extern __shared__ unsigned smem[];
  unsigned* sacc = smem;                       // RT * CP u32 accumulators
  int*      slab = (int*)(smem + RT * CP);     // cpb staged labels

  const int tid   = threadIdx.x;
  const int col0  = blockIdx.x * cpb;
  const int ncols = min(cpb, N - col0);
  const int row0  = blockIdx.y * RT;
  const int nrows = min(RT, B - row0);

  for (int i = tid; i < RT * CP; i += TPB) sacc[i] = 0u;

  // ---- stage labels chunk into LDS (CDNA5 async global->LDS if available) --
#if HAVE_ASYNC_LDS
  for (int j = tid; j < ncols; j += TPB) {
    __builtin_amdgcn_global_load_async_to_lds_b32(
        (AS1 int*)(uintptr_t)(labels + col0 + j),
        (AS3 int*)(uint32_t)(uintptr_t)(slab + j),
        0 /*offset*/, 0 /*cpol*/);
  }
  __builtin_amdgcn_s_wait_asynccnt(0);
#else
  for (int j = tid; j < ncols; j += TPB) slab[j] = labels[col0 + j];
#endif
  __syncthreads();

  const size_t rowbase = (size_t)row0 * (size_t)N + (size_t)col0;

  if (((N & 3) == 0) && ((col0 & 3) == 0)) {
    // vector path: 4 columns per thread per step, b128 NT loads, 16 in flight
    const int j4 = ncols & ~3;
    for (int j = tid * 4; j < j4; j += TPB * 4) {
      v4i lab = *(const v4i*)(slab + j);       // ds_load_b128
      const float* p = values + rowbase + j;
#pragma unroll
      for (int r = 0; r < RT; ++r) {
        if (r < nrows) {
          v4f v = __builtin_nontemporal_load((const v4f*)(p + (size_t)r * N));
          unsigned* a = sacc + r * CP;
          atomicMax(a + lab.x, enc_f32(v.x));  // ds_max_u32 (no return)
          atomicMax(a + lab.y, enc_f32(v.y));
          atomicMax(a + lab.z, enc_f32(v.z));
          atomicMax(a + lab.w, enc_f32(v.w));
        }
      }
    }
    for (int j = j4 + tid; j < ncols; j += TPB) {       // tail columns
      int c = slab[j];
      for (int r = 0; r < nrows; ++r) {
        float v = __builtin_nontemporal_load(values + rowbase + (size_t)r * N + j);
        atomicMax(sacc + r * CP + c, enc_f32(v));
      }
    }
  } else {
    // generic scalar path (still coalesced across lanes)
    for (int j = tid; j < ncols; j += TPB) {
      int c = slab[j];
      for (int r = 0; r < nrows; ++r) {
        float v = __builtin_nontemporal_load(values + rowbase + (size_t)r * N + j);
        atomicMax(sacc + r * CP + c, enc_f32(v));
      }
    }
  }

  __syncthreads();

  // ---- merge per-block partials into global accumulator ---------------------
  for (int i = tid; i < nrows * C; i += TPB) {
    int r = i / C;
    int c = i - r * C;
    unsigned v = sacc[r * CP + c];
    if (v) atomicMax(acc + (size_t)(row0 + r) * C + c, v);  // global_atomic_max_u32
  }
}

__global__ void __launch_bounds__(TPB)
strat_final_kernel(const unsigned* __restrict__ acc,
                   float* __restrict__ out, int n) {
  int i = blockIdx.x * TPB + threadIdx.x;
  if (i < n) {
    unsigned e = acc[i];
    out[i] = e ? dec_u32(e) : __uint_as_float(0xFF800000u);  // empty -> -inf
  }
}

extern "C" void kernel_launch(void* const* d_in, const int* in_sizes, int n_in,
                              void* d_out, int out_size, void* d_ws, size_t ws_size,
                              hipStream_t stream) {
  const float* values = (const float*)d_in[0];
  const int*   labels = (const int*)d_in[1];
  // d_in[2] is the num_classes scalar on device; C is recovered from sizes.

  const int N = in_sizes[1];
  const int B = (N > 0) ? (in_sizes[0] / N) : 0;
  const int C = (B > 0) ? (out_size / B) : 0;
  if (N <= 0 || B <= 0 || C <= 0) return;

  unsigned* acc = (unsigned*)d_ws;           // B*C u32 encoded accumulators
  float*    out = (float*)d_out;
  const int bc  = B * C;

  // 1) zero encoded accumulators
  strat_init_kernel<<<(bc + TPB - 1) / TPB, TPB, 0, stream>>>(acc, bc);

  // 2) main streaming reduction
  int gx = 128;                              // column chunks
  if (gx > (N + 3) / 4) gx = (N + 3) / 4;
  if (gx < 1) gx = 1;
  int cpb = (((N + gx - 1) / gx) + 3) & ~3;  // columns per block, mult of 4
  gx = (N + cpb - 1) / cpb;
  const int gy = (B + RT - 1) / RT;          // row tiles
  const int CP = C | 1;                      // odd stride vs 64 LDS banks
  const size_t lds_bytes = (size_t)(RT * CP + cpb) * sizeof(unsigned);

  strat_main_kernel<<<dim3(gx, gy), TPB, lds_bytes, stream>>>(
      values, labels, acc, N, B, C, CP, cpb);

  // 3) decode u32 -> f32
  strat_final_kernel<<<(bc + TPB - 1) / TPB, TPB, 0, stream>>>(acc, out, bc);
}